// GraphAutoEncoder_42941083025547
// MI455X (gfx1250) — compile-verified
//
#include <hip/hip_runtime.h>
#include <stdint.h>

typedef __attribute__((ext_vector_type(2))) float v2f;
typedef __attribute__((ext_vector_type(8))) float v8f;

#define N_NODES    100000
#define N_EDGES    3200000
#define IN_DIM     512
#define LATENT_DIM 128

// CDNA5 async global->LDS copy path (ASYNCcnt-tracked), compile-safe guarded.
#if defined(__HIP_DEVICE_COMPILE__) && __has_builtin(__builtin_amdgcn_global_load_async_to_lds_b128)
#define HAVE_ASYNC_LDS 1
#else
#define HAVE_ASYNC_LDS 0
#endif

#if HAVE_ASYNC_LDS
// Builtin signature (from hipcc diagnostic): (v4i AS1*, v4i AS3*, imm, imm)
typedef int v4i_vs __attribute__((vector_size(4 * sizeof(int))));
typedef __attribute__((address_space(1))) v4i_vs* g_v4i_ptr;
typedef __attribute__((address_space(3))) v4i_vs* l_v4i_ptr;
#endif

// ---------------------------------------------------------------------------
// Zero-init (d_out is poisoned by the harness and we accumulate with atomics)
// ---------------------------------------------------------------------------
__global__ __launch_bounds__(256) void zero_f4_kernel(float4* __restrict__ p, long n4) {
  long i = (long)blockIdx.x * blockDim.x + threadIdx.x;
  if (i < n4) p[i] = make_float4(0.f, 0.f, 0.f, 0.f);
}

// ---------------------------------------------------------------------------
// Dense GEMM via V_WMMA_F32_16X16X4_F32 (exact f32 path, wave32).
//   C[16 rows x 128 cols] tile per block; 8 waves -> 8 x 16-col tiles.
//   A (16 x K) staged in LDS with +4 float row pad to spread ds banks;
//   staging uses GLOBAL_LOAD_ASYNC_TO_LDS_B128 when the toolchain exposes it.
//   Optional row gather (perm) for the corrupted-view encoder pass.
// A-frag layout (32-bit A 16x4): lanes 0-15 row M=l, VGPR0=K+0,VGPR1=K+1;
//                                lanes 16-31 row M=l-16, VGPR0=K+2,VGPR1=K+3.
// B-frag layout mirrors it over N: VGPR0 = row K (+2 for hi half), N = lane%16.
// C layout: VGPR v -> M = v + 8*(lane/16), N = lane%16.
// ---------------------------------------------------------------------------
template <int K, bool GATHER>
__global__ __launch_bounds__(256)
void gemm_wmma_f32(const float* __restrict__ A, const float* __restrict__ B,
                   float* __restrict__ C, const int* __restrict__ perm, int ncols) {
  constexpr int LDA  = K + 4;        // padded LDS row stride (floats)
  constexpr int K4   = K / 4;
  constexpr int LDA4 = LDA / 4;
  extern __shared__ float smem[];    // 16 * LDA floats

  const int row0    = blockIdx.x * 16;
  const int colBase = blockIdx.y * 128;

  // ---- stage 16 x K A-tile into LDS, coalesced float4 ----
  float4*       s4 = (float4*)smem;
  const float4* A4 = (const float4*)A;
  for (int i = threadIdx.x; i < 16 * K4; i += 256) {
    const int r = i / K4, c = i % K4;
    const int src = GATHER ? perm[row0 + r] : (row0 + r);
#if HAVE_ASYNC_LDS
    __builtin_amdgcn_global_load_async_to_lds_b128(
        (g_v4i_ptr)(uintptr_t)&A4[(long)src * K4 + c],
        (l_v4i_ptr)(uint32_t)(uintptr_t)&s4[r * LDA4 + c],
        /*offset=*/0, /*cpol=*/0);
#else
    s4[r * LDA4 + c] = A4[(long)src * K4 + c];
#endif
  }
#if HAVE_ASYNC_LDS
#if __has_builtin(__builtin_amdgcn_s_wait_asynccnt)
  __builtin_amdgcn_s_wait_asynccnt(0);
#else
  asm volatile("s_wait_asynccnt 0x0" ::: "memory");
#endif
#endif
  __syncthreads();

  const int wave = threadIdx.x >> 5;
  const int lane = threadIdx.x & 31;
  const int hi   = lane >> 4;        // 0 = lanes 0-15, 1 = lanes 16-31
  const int l15  = lane & 15;
  const int c0   = colBase + wave * 16;

  const float* As = smem + l15 * LDA + 2 * hi;               // A row l15, K offset 2*hi
  const float* Bp = B + (long)(2 * hi) * ncols + c0 + l15;   // B row 2*hi, col c0+l15

  v8f acc = {};
#pragma unroll 8
  for (int k = 0; k < K; k += 4) {
    v2f a = *reinterpret_cast<const v2f*>(As + k);           // ds_load_b64 (2addr-paired)
    v2f b;
    b.x = Bp[(long)k * ncols];
    b.y = Bp[(long)(k + 1) * ncols];
    acc = __builtin_amdgcn_wmma_f32_16x16x4_f32(
        /*neg_a=*/false, a, /*neg_b=*/false, b,
        /*c_mod=*/(short)0, acc, /*reuse_a=*/false, /*reuse_b=*/false);
  }

  float* Cp = C + (long)(row0 + 8 * hi) * ncols + c0 + l15;
#pragma unroll
  for (int v = 0; v < 8; ++v) Cp[(long)v * ncols] = acc[v];
}

// ---------------------------------------------------------------------------
// SpMM scatter: one wave32 per edge. 32 lanes x float4 = 128 dims per pass.
// out[row[e]] += vals[e] * Z[col[e]]  via global_atomic_add_f32.
// Z rows are L2-resident (z: 51 MB, zd: 205 MB vs 192 MB L2), so the binding
// resource is L2 atomic throughput — structure chosen to maximize it.
// ---------------------------------------------------------------------------
template <int D>
__global__ __launch_bounds__(256)
void spmm_scatter(const int* __restrict__ row, const int* __restrict__ col,
                  const float* __restrict__ vals, const float* __restrict__ Z,
                  float* __restrict__ Out, int nEdges) {
  const int wave = threadIdx.x >> 5;
  const int lane = threadIdx.x & 31;
  const int e = blockIdx.x * 8 + wave;
  if (e >= nEdges) return;

  const int   r = row[e];
  const int   c = col[e];
  const float v = vals[e];

  const float4* zr   = (const float4*)(Z + (long)c * D);
  float*        orow = Out + (long)r * D;

#pragma unroll
  for (int j = 0; j < D / 128; ++j) {
    const float4 zv = zr[j * 32 + lane];
    const int base  = (j * 32 + lane) * 4;
    unsafeAtomicAdd(orow + base + 0, v * zv.x);
    unsafeAtomicAdd(orow + base + 1, v * zv.y);
    unsafeAtomicAdd(orow + base + 2, v * zv.z);
    unsafeAtomicAdd(orow + base + 3, v * zv.w);
  }
}

// ---------------------------------------------------------------------------
// Orchestration:
//   zero(d_out)
//   z  = x @ W_enc            (ws)          -> h   = spmm(z)   (d_out)
//   zd = h @ W_dec            (ws, reused)  -> out = spmm(zd)  (d_out)
//   za = x[perm] @ W_enc      (ws, reused)  -> h_a = spmm(za)  (d_out)
// ---------------------------------------------------------------------------
extern "C" void kernel_launch(void* const* d_in, const int* in_sizes, int n_in,
                              void* d_out, int out_size, void* d_ws, size_t ws_size,
                              hipStream_t stream) {
  (void)in_sizes; (void)n_in; (void)out_size; (void)ws_size;

  const float* x     = (const float*)d_in[0];
  const float* W_enc = (const float*)d_in[1];
  const float* W_dec = (const float*)d_in[2];
  const float* vals  = (const float*)d_in[3];
  const int*   row   = (const int*)d_in[4];
  const int*   col   = (const int*)d_in[5];
  const int*   perm  = (const int*)d_in[6];

  float* out = (float*)d_out;                        // [N, 512]
  float* h   = out + (long)N_NODES * IN_DIM;         // [N, 128]
  float* h_a = h + (long)N_NODES * LATENT_DIM;       // [N, 128]
  float* ws  = (float*)d_ws;                         // needs N*512 floats (205 MB)

  // 0) zero the atomically-accumulated output buffer
  const long n4 = (long)N_NODES * (IN_DIM + 2 * LATENT_DIM) / 4;
  zero_f4_kernel<<<(int)((n4 + 255) / 256), 256, 0, stream>>>((float4*)d_out, n4);

  const dim3   gridEnc(N_NODES / 16, LATENT_DIM / 128);
  const dim3   gridDec(N_NODES / 16, IN_DIM / 128);
  const size_t shEnc = 16 * (IN_DIM + 4) * sizeof(float);
  const size_t shDec = 16 * (LATENT_DIM + 4) * sizeof(float);
  const int    spmmBlocks = (N_EDGES + 7) / 8;

  // 1) z = x @ W_enc
  gemm_wmma_f32<IN_DIM, false><<<gridEnc, 256, shEnc, stream>>>(x, W_enc, ws, nullptr, LATENT_DIM);
  // 2) h = adj @ z
  spmm_scatter<LATENT_DIM><<<spmmBlocks, 256, 0, stream>>>(row, col, vals, ws, h, N_EDGES);
  // 3) zd = h @ W_dec   (ws reused; h is complete — same-stream ordering)
  gemm_wmma_f32<LATENT_DIM, false><<<gridDec, 256, shDec, stream>>>(h, W_dec, ws, nullptr, IN_DIM);
  // 4) out = adj @ zd
  spmm_scatter<IN_DIM><<<spmmBlocks, 256, 0, stream>>>(row, col, vals, ws, out, N_EDGES);
  // 5) za = x[perm] @ W_enc  (ws reused)
  gemm_wmma_f32<IN_DIM, true><<<gridEnc, 256, shEnc, stream>>>(x, W_enc, ws, perm, LATENT_DIM);
  // 6) h_a = adj @ za
  spmm_scatter<LATENT_DIM><<<spmmBlocks, 256, 0, stream>>>(row, col, vals, ws, h_a, N_EDGES);
}